// MtpDrafterLayerANE_67242007987135
// MI455X (gfx1250) — compile-verified
//
#include <hip/hip_runtime.h>
#include <math.h>

#define D_   2560
#define NH_  8
#define NKV_ 4
#define HD_  256
#define FFN_ 10240
#define QD_  2048
#define CTX_ 4096
#define EPS_ 1e-6f

typedef __attribute__((ext_vector_type(2))) float v2f;
typedef __attribute__((ext_vector_type(8))) float v8f;
typedef __attribute__((ext_vector_type(4))) unsigned int u32x4;
typedef __attribute__((ext_vector_type(8))) int          i32x8;
typedef __attribute__((ext_vector_type(4))) int          i32x4;

#if defined(__HIP_DEVICE_COMPILE__) && \
    __has_builtin(__builtin_amdgcn_tensor_load_to_lds) && \
    __has_builtin(__builtin_amdgcn_s_wait_tensorcnt)
#define USE_TDM 1
#else
#define USE_TDM 0
#endif

// LDS offset of a __shared__ object: flat LDS-aperture address, low 32 bits
// are the byte offset within the workgroup's LDS allocation.
__device__ __forceinline__ unsigned lds_byte_off(const void* p) {
  return (unsigned)(unsigned long long)p;
}

#if USE_TDM
// Issue one TDM descriptor: copy a 128-row x 32-float tile (row stride ldw
// floats) from global 'gsrc' into LDS at byte offset 'ldsoff', padding the
// LDS destination by 2 dwords after every 32 dwords (-> row stride 34 floats,
// bank-conflict-free for the WMMA B reads). Tracked by TENSORcnt.
__device__ __forceinline__ void tdm_load_tile(const float* gsrc, unsigned ldsoff,
                                              int ldw) {
  const unsigned long long ga = (unsigned long long)gsrc;
  u32x4 g0;
  g0[0] = 1u;                                     // count=1, user mode, no gather
  g0[1] = ldsoff;                                 // lds_addr (bytes)
  g0[2] = (unsigned)ga;                           // global_addr[31:0]
  g0[3] = (unsigned)((ga >> 32) & 0x01FFFFFFu)    // global_addr[56:32]
        | (2u << 30);                             // type = 2 ("image")
  i32x8 g1;
  g1[0] = (int)((2u << 16)                        // data_size = 4 bytes
              | (1u << 20)                        // pad_enable
              | (4u << 22)                        // pad_interval: 32 dwords
              | (1u << 25));                      // pad_amount: 2 dwords
  g1[1] = (int)(32u << 16);                       // tensor_dim0 = 32 (lo16 @63:48)
  g1[2] = (int)(128u << 16);                      // dim0 hi16=0, tensor_dim1=128 lo16
  g1[3] = (int)(32u << 16);                       // dim1 hi16=0, tile_dim0 = 32
  g1[4] = (int)128u;                              // tile_dim1 = 128, tile_dim2 = 0
  g1[5] = (int)(unsigned)ldw;                     // tensor_dim0_stride[31:0] (elems)
  g1[6] = 0;                                      // stride0 hi16, stride1 lo16
  g1[7] = 0;                                      // stride1 hi32
  const i32x4 gz  = {0, 0, 0, 0};                 // groups 2/3: 2-D tensor
  const i32x8 gz8 = {0, 0, 0, 0, 0, 0, 0, 0};     // extra arg (clang-23 form)
  __builtin_amdgcn_tensor_load_to_lds(g0, g1, gz, gz, gz8, 0);
}
#endif

// ---------------------------------------------------------------------------
// WMMA GEMV:  y[b][r][n] = sum_k x[b][r][k] * W[b][n][k]   (r < na, na<=2)
// Block = 256 threads = 8 waves; each block produces 128 output rows (n).
// Weight tiles (128 rows x 32 K) are streamed into LDS by the Tensor Data
// Mover (double-buffered, overlapped with compute) and fed to
// V_WMMA_F32_16X16X4_F32 in the B (4x16) layout. The activation vector(s)
// sit in A rows 0..na-1. Requires K % 32 == 0, N % 128 == 0.
// ---------------------------------------------------------------------------
__global__ __launch_bounds__(256) void gemv_wmma_kernel(
    const float* __restrict__ W, const float* __restrict__ x,
    float* __restrict__ y,
    int K, int ldw, int na, int ldx, int ldy,
    long wBatch, long xBatch, long yBatch)
{
  constexpr int KSTEP = 32;
  constexpr int LDP   = 34;              // padded row stride (TDM pad / manual)
  __shared__ float tileW[2][128 * LDP];
  __shared__ float xs[2 * KSTEP];

  const int b = blockIdx.y;
  W += (size_t)b * wBatch;
  x += (size_t)b * xBatch;
  y += (size_t)b * yBatch;

  const int tid   = threadIdx.x;
  const int lane  = tid & 31;
  const int wave  = tid >> 5;
  const int m     = lane & 15;           // A-row / B-column index for this lane
  const int khalf = (lane >> 4) << 1;    // lanes 16..31 hold K+2,K+3
  const int rowBase = blockIdx.x * 128;

  v8f c = {};                            // fp32 accumulator (C/D 16x16)
  int cur = 0;

#if USE_TDM
  if (wave == 0)
    tdm_load_tile(W + (size_t)rowBase * ldw, lds_byte_off(&tileW[0][0]), ldw);
#else
  const int r0 = tid >> 3;               // 32 rows per load pass
  const int c0 = (tid & 7) << 2;         // float4 column within 32-wide tile
#endif

  for (int kb = 0; kb < K; kb += KSTEP) {
    // ---- stage activation slice(s) (tiny, synchronous) ----
    if (tid < na * KSTEP) {
      const int r = tid >> 5, cc = tid & 31;
      xs[r * KSTEP + cc] = x[r * ldx + kb + cc];
    }
#if USE_TDM
    // ---- issue next tile's DMA, then wait for the current tile ----
    if (wave == 0) {
      if (kb + KSTEP < K) {
        tdm_load_tile(W + (size_t)rowBase * ldw + (kb + KSTEP),
                      lds_byte_off(&tileW[cur ^ 1][0]), ldw);
        __builtin_amdgcn_s_wait_tensorcnt(1);   // in-order: current tile landed
      } else {
        __builtin_amdgcn_s_wait_tensorcnt(0);
      }
    }
#else
    // ---- fallback: cooperative float4 global->LDS staging ----
    #pragma unroll
    for (int p = 0; p < 4; ++p) {
      const int row = p * 32 + r0;
      const float4 v = *reinterpret_cast<const float4*>(
          W + (size_t)(rowBase + row) * ldw + kb + c0);
      float* dst = &tileW[cur][row * LDP + c0];
      dst[0] = v.x; dst[1] = v.y; dst[2] = v.z; dst[3] = v.w;
    }
    if (kb + KSTEP < K)
      __builtin_prefetch(W + (size_t)(rowBase + r0) * ldw + kb + KSTEP, 0, 1);
#endif
    __syncthreads();

    // ---- 8 WMMAs consume the 32-wide K slab ----
    const float* bp0 = &tileW[cur][(wave * 16 + m) * LDP + khalf];
    const bool  hasA = (m < na);
    const int   ar   = hasA ? m : 0;
    const float* ap0 = &xs[ar * KSTEP + khalf];
    #pragma unroll
    for (int kk = 0; kk < KSTEP; kk += 4) {
      v2f av, bv;
      bv.x = bp0[kk];
      bv.y = bp0[kk + 1];
      const float a0 = ap0[kk], a1 = ap0[kk + 1];
      av.x = hasA ? a0 : 0.f;
      av.y = hasA ? a1 : 0.f;
      c = __builtin_amdgcn_wmma_f32_16x16x4_f32(
              false, av, false, bv, (short)0, c, false, false);
    }
    __syncthreads();
    cur ^= 1;
  }

  // D row r lives in C VGPR r, lanes 0..15 (N = lane)
  if (lane < 16) {
    const int n = rowBase + wave * 16 + lane;
    y[n] = c[0];
    if (na > 1) y[(size_t)ldy + n] = c[1];
  }
}

// ---------------------------------------------------------------------------
// out = (res ? res + : ) _rms(in) * (1 + w)      (single block, n <= 2560)
// ---------------------------------------------------------------------------
__global__ __launch_bounds__(256) void rmsnorm_kernel(
    const float* __restrict__ in, const float* __restrict__ w,
    const float* __restrict__ res, float* __restrict__ out, int n)
{
  __shared__ float red[256];
  __shared__ float s_rs;
  const int t = threadIdx.x;
  float ss = 0.f;
  for (int i = t; i < n; i += 256) { const float v = in[i]; ss += v * v; }
  red[t] = ss; __syncthreads();
  for (int s = 128; s > 0; s >>= 1) {
    if (t < s) red[t] += red[t + s];
    __syncthreads();
  }
  if (t == 0) s_rs = rsqrtf(red[0] * (1.f / n) + EPS_);
  __syncthreads();
  const float rs = s_rs;
  for (int i = t; i < n; i += 256) {
    const float v = in[i] * rs * (1.f + w[i]);
    out[i] = res ? res[i] + v : v;
  }
}

// ---------------------------------------------------------------------------
// per-head q RMS (no weight) + RoPE.  grid = NH_, block = HD_ (256)
// ---------------------------------------------------------------------------
__global__ __launch_bounds__(256) void qnorm_rope_kernel(
    const float* __restrict__ q, const float* __restrict__ cosv,
    const float* __restrict__ sinv, float* __restrict__ out)
{
  __shared__ float red[256];
  __shared__ float qn[256];
  __shared__ float s_rs;
  const int h = blockIdx.x, d = threadIdx.x;
  const float v = q[h * HD_ + d];
  red[d] = v * v; __syncthreads();
  for (int s = 128; s > 0; s >>= 1) {
    if (d < s) red[d] += red[d + s];
    __syncthreads();
  }
  if (d == 0) s_rs = rsqrtf(red[0] * (1.f / HD_) + EPS_);
  __syncthreads();
  const float nv = v * s_rs;
  qn[d] = nv; __syncthreads();
  const float rot = (d < HD_ / 2) ? -qn[d + HD_ / 2] : qn[d - HD_ / 2];
  out[h * HD_ + d] = nv * cosv[d] + rot * sinv[d];
}

// ---------------------------------------------------------------------------
// in-place masked softmax over CTX per head.  grid = NH_
// ---------------------------------------------------------------------------
__global__ __launch_bounds__(256) void softmax_kernel(
    float* __restrict__ sc, const float* __restrict__ mask)
{
  __shared__ float red[256];
  __shared__ float sh;
  const int h = blockIdx.x, t = threadIdx.x;
  float* row = sc + (size_t)h * CTX_;
  float mx = -3.4e38f;
  for (int j = t; j < CTX_; j += 256) mx = fmaxf(mx, row[j] + mask[j]);
  red[t] = mx; __syncthreads();
  for (int s = 128; s > 0; s >>= 1) {
    if (t < s) red[t] = fmaxf(red[t], red[t + s]);
    __syncthreads();
  }
  if (t == 0) sh = red[0];
  __syncthreads();
  mx = sh;
  float sum = 0.f;
  for (int j = t; j < CTX_; j += 256) {
    const float e = expf(row[j] + mask[j] - mx);
    row[j] = e;
    sum += e;
  }
  red[t] = sum; __syncthreads();
  for (int s = 128; s > 0; s >>= 1) {
    if (t < s) red[t] += red[t + s];
    __syncthreads();
  }
  if (t == 0) sh = red[0];
  __syncthreads();
  const float inv = 1.f / sh;
  for (int j = t; j < CTX_; j += 256) row[j] *= inv;
}

// ---------------------------------------------------------------------------
// attn @ V, fully coalesced (thread per output column d).
// grid = (CTX_/256 chunks, NH_ heads), block = HD_ (256)
// ---------------------------------------------------------------------------
__global__ __launch_bounds__(256) void attn_v_partial_kernel(
    const float* __restrict__ attn, const float* __restrict__ V,
    float* __restrict__ part)
{
  __shared__ float aw[256];
  const int ck = blockIdx.x, h = blockIdx.y, d = threadIdx.x;
  const int g = h >> 1;                  // GQA: 2 q heads per KV head
  const int j0 = ck * 256;
  aw[d] = attn[(size_t)h * CTX_ + j0 + d];
  __syncthreads();
  const float* vp = V + ((size_t)g * CTX_ + j0) * HD_ + d;
  float acc = 0.f;
  #pragma unroll 4
  for (int j = 0; j < 256; ++j) acc += aw[j] * vp[(size_t)j * HD_];
  part[((h * (CTX_ / 256) + ck) * HD_) + d] = acc;
}

__global__ __launch_bounds__(256) void attn_v_reduce_kernel(
    const float* __restrict__ part, float* __restrict__ a)
{
  const int h = blockIdx.x, d = threadIdx.x;
  float s = 0.f;
  #pragma unroll
  for (int ck = 0; ck < CTX_ / 256; ++ck)
    s += part[(h * (CTX_ / 256) + ck) * HD_ + d];
  a[h * HD_ + d] = s;                    // head-major concat == reference reshape
}

// ---------------------------------------------------------------------------
// GeGLU: o = gelu_tanh(g) * u
// ---------------------------------------------------------------------------
__global__ __launch_bounds__(256) void geglu_kernel(
    const float* __restrict__ g, const float* __restrict__ u,
    float* __restrict__ o, int n)
{
  const int i = blockIdx.x * 256 + threadIdx.x;
  if (i < n) {
    const float xv = g[i];
    const float t = tanhf(0.7978845608028654f * (xv + 0.044715f * xv * xv * xv));
    o[i] = 0.5f * xv * (1.f + t) * u[i];
  }
}

// ---------------------------------------------------------------------------
extern "C" void kernel_launch(void* const* d_in, const int* in_sizes, int n_in,
                              void* d_out, int out_size, void* d_ws, size_t ws_size,
                              hipStream_t stream)
{
  (void)in_sizes; (void)n_in; (void)out_size; (void)ws_size;

  const float* x        = (const float*)d_in[0];
  const float* cosv     = (const float*)d_in[1];
  const float* sinv     = (const float*)d_in[2];
  const float* mask     = (const float*)d_in[3];
  const float* cache_k  = (const float*)d_in[4];
  const float* cache_v  = (const float*)d_in[5];
  /* d_in[6] = ctx_len (== CTX_) */
  const float* w_in     = (const float*)d_in[7];
  const float* w_pattn  = (const float*)d_in[8];
  const float* w_preffn = (const float*)d_in[9];
  const float* w_pffn   = (const float*)d_in[10];
  const float* Wq       = (const float*)d_in[11];
  const float* Wo       = (const float*)d_in[12];
  const float* Wg       = (const float*)d_in[13];
  const float* Wu       = (const float*)d_in[14];
  const float* Wd       = (const float*)d_in[15];
  float* out = (float*)d_out;

  float* ws    = (float*)d_ws;
  float* h     = ws;              // D_
  float* qr    = h    + D_;       // QD_
  float* qn    = qr   + QD_;      // QD_
  float* sc    = qn   + QD_;      // NH_*CTX_
  float* part  = sc   + NH_*CTX_; // NH_*(CTX_/256)*HD_
  float* av    = part + NH_*(CTX_/256)*HD_;  // QD_
  float* aproj = av   + QD_;      // D_
  float* x2    = aproj+ D_;       // D_
  float* h2    = x2   + D_;       // D_
  float* gb    = h2   + D_;       // FFN_
  float* ub    = gb   + FFN_;     // FFN_
  float* fi    = ub   + FFN_;     // FFN_
  float* fo    = fi   + FFN_;     // D_

  // 1) h = rmsnorm(x, w_in_ln)
  rmsnorm_kernel<<<1, 256, 0, stream>>>(x, w_in, nullptr, h, D_);
  // 2) q = Wq @ h
  gemv_wmma_kernel<<<dim3(QD_/128, 1), 256, 0, stream>>>(
      Wq, h, qr, D_, D_, 1, D_, 0, 0, 0, 0);
  // 3) per-head q RMS + RoPE
  qnorm_rope_kernel<<<NH_, HD_, 0, stream>>>(qr, cosv, sinv, qn);
  // 4) scores: one pass over K-cache computes both q heads of each KV group
  gemv_wmma_kernel<<<dim3(CTX_/128, NKV_), 256, 0, stream>>>(
      cache_k, qn, sc, HD_, HD_, 2, HD_, CTX_,
      (long)CTX_ * HD_, (long)2 * HD_, (long)2 * CTX_);
  // 5) softmax(+mask) per head, in place
  softmax_kernel<<<NH_, 256, 0, stream>>>(sc, mask);
  // 6) attn @ V (split over ctx chunks), then reduce
  attn_v_partial_kernel<<<dim3(CTX_/256, NH_), HD_, 0, stream>>>(sc, cache_v, part);
  attn_v_reduce_kernel<<<NH_, HD_, 0, stream>>>(part, av);
  // 7) o-proj
  gemv_wmma_kernel<<<dim3(D_/128, 1), 256, 0, stream>>>(
      Wo, av, aproj, QD_, QD_, 1, QD_, 0, 0, 0, 0);
  // 8) x2 = x + rmsnorm(aproj, w_post_attn)
  rmsnorm_kernel<<<1, 256, 0, stream>>>(aproj, w_pattn, x, x2, D_);
  // 9) h2 = rmsnorm(x2, w_pre_ffn)
  rmsnorm_kernel<<<1, 256, 0, stream>>>(x2, w_preffn, nullptr, h2, D_);
  // 10) gate / up projections
  gemv_wmma_kernel<<<dim3(FFN_/128, 1), 256, 0, stream>>>(
      Wg, h2, gb, D_, D_, 1, D_, 0, 0, 0, 0);
  gemv_wmma_kernel<<<dim3(FFN_/128, 1), 256, 0, stream>>>(
      Wu, h2, ub, D_, D_, 1, D_, 0, 0, 0, 0);
  // 11) GeGLU
  geglu_kernel<<<FFN_/256, 256, 0, stream>>>(gb, ub, fi, FFN_);
  // 12) down projection
  gemv_wmma_kernel<<<dim3(D_/128, 1), 256, 0, stream>>>(
      Wd, fi, fo, FFN_, FFN_, 1, FFN_, 0, 0, 0, 0);
  // 13) out = x2 + rmsnorm(fo, w_post_ffn)
  rmsnorm_kernel<<<1, 256, 0, stream>>>(fo, w_pffn, x2, out, D_);
}